// GraphEmbedding_57878979281306
// MI455X (gfx1250) — compile-verified
//
#include <hip/hip_runtime.h>

#define D 64

typedef float v2f __attribute__((ext_vector_type(2)));
typedef float v8f __attribute__((ext_vector_type(8)));

// ---------- small utility kernels ----------
__global__ void k_fill(float* __restrict__ p, float v, int n) {
  int i = blockIdx.x * blockDim.x + threadIdx.x;
  if (i < n) p[i] = v;
}

__global__ void k_deg_accum(const int* __restrict__ dst, float* __restrict__ deg, int nE) {
  int e = blockIdx.x * blockDim.x + threadIdx.x;
  if (e < nE) atomicAdd(&deg[dst[e]], 1.0f);
}

__global__ void k_rsqrt_inplace(float* __restrict__ p, int n) {
  int i = blockIdx.x * blockDim.x + threadIdx.x;
  if (i < n) p[i] = rsqrtf(p[i]);
}

// ---------- dense transform: C[M x 64] = A[M x 64] * B[64 x 64], fp32 WMMA ----------
// Block = 256 threads = 8 waves; each block covers a 32-row x 64-col chunk of C.
// Wave w: row-tile = (w>>2), col-tile = (w&3). K is walked in 16 steps of 4.
__global__ void __launch_bounds__(256) k_gemm_wmma(const float* __restrict__ A,
                                                   const float* __restrict__ B,
                                                   float* __restrict__ C, int M) {
  __shared__ float Bs[D * D];           // 16 KB of the 320 KB/WGP LDS
  int tid = threadIdx.x;
  for (int i = tid; i < D * D; i += 256) Bs[i] = B[i];
  __syncthreads();

  int wave   = tid >> 5;
  int lane   = tid & 31;
  int lane16 = lane & 15;
  int khalf  = lane >> 4;               // half-wave splits the K=4 slice into {0,1} / {2,3}

  int rowTile = (blockIdx.x * 2 + (wave >> 2)) * 16;
  int colTile = (wave & 3) * 16;
  if (rowTile >= M) return;             // (no barriers past this point)

  const float* Arow = A + (size_t)(rowTile + lane16) * D;
  v8f acc = {};
#pragma unroll
  for (int kk = 0; kk < 16; ++kk) {
    int k = kk * 4 + khalf * 2;
    v2f a, b;
    // A 16x4 fragment: lane = M, components = consecutive K, half-wave = K offset +2
    a.x = Arow[k];
    a.y = Arow[k + 1];
    // B 4x16 fragment: lane = N, components = consecutive K, half-wave = K offset +2
    b.x = Bs[(k + 0) * D + colTile + lane16];
    b.y = Bs[(k + 1) * D + colTile + lane16];
    acc = __builtin_amdgcn_wmma_f32_16x16x4_f32(false, a, false, b,
                                                (short)0, acc, false, false);
  }

  // C/D layout: VGPR r -> row (rowTile + khalf*8 + r), col = colTile + lane16
  int    col   = colTile + lane16;
  size_t rbase = (size_t)rowTile + (size_t)khalf * 8;
#pragma unroll
  for (int r = 0; r < 8; ++r)
    C[(rbase + r) * D + col] = acc[r];
}

// ---------- aggregation ----------
// agg[i] = hW[i] * dinv[i]^2   (self-loop term; also fuses the buffer zeroing)
__global__ void k_selfloop(const float* __restrict__ h, const float* __restrict__ dinv,
                           float* __restrict__ agg, int total) {
  int idx = blockIdx.x * blockDim.x + threadIdx.x;
  if (idx < total) {
    int i = idx >> 6;
    float dv = dinv[i];
    agg[idx] = h[idx] * dv * dv;
  }
}

// 64 threads per edge: coalesced gather of the 256 B source row, f32 atomic scatter.
__global__ void k_edge_scatter(const float* __restrict__ h, const float* __restrict__ dinv,
                               const int* __restrict__ src, const int* __restrict__ dst,
                               float* __restrict__ agg, int nE) {
  int idx = blockIdx.x * blockDim.x + threadIdx.x;   // nE*64 = 102.4M < 2^31
  int e = idx >> 6;
  int c = idx & 63;
  if (e < nE) {
    int s = src[e];
    int d = dst[e];
    float norm = dinv[s] * dinv[d];
    atomicAdd(&agg[(size_t)d * D + c], h[(size_t)s * D + c] * norm);
  }
}

__global__ void k_bias_act(float* __restrict__ h, const float* __restrict__ b,
                           int total, int relu) {
  int idx = blockIdx.x * blockDim.x + threadIdx.x;
  if (idx < total) {
    float v = h[idx] + b[idx & 63];
    h[idx] = relu ? fmaxf(v, 0.0f) : v;
  }
}

// ---------- mean pool over 16 graphs ----------
__global__ void k_pool_accum(const float* __restrict__ h, const int* __restrict__ batch,
                             float* __restrict__ sums, float* __restrict__ cnts, int total) {
  int idx = blockIdx.x * blockDim.x + threadIdx.x;
  if (idx < total) {
    int i = idx >> 6;
    int c = idx & 63;
    int g = batch[i];
    atomicAdd(&sums[(size_t)g * D + c], h[idx]);
    if (c == 0) atomicAdd(&cnts[g], 1.0f);
  }
}

__global__ void k_pool_div(const float* __restrict__ sums, const float* __restrict__ cnts,
                           float* __restrict__ out, int total) {
  int idx = blockIdx.x * blockDim.x + threadIdx.x;
  if (idx < total) out[idx] = sums[idx] / fmaxf(cnts[idx >> 6], 1.0f);
}

// ---------- launcher ----------
extern "C" void kernel_launch(void* const* d_in, const int* in_sizes, int n_in,
                              void* d_out, int out_size, void* d_ws, size_t ws_size,
                              hipStream_t stream) {
  const float* x     = (const float*)d_in[0];
  const float* W1    = (const float*)d_in[1];
  const float* b1    = (const float*)d_in[2];
  const float* W2    = (const float*)d_in[3];
  const float* b2    = (const float*)d_in[4];
  const int*   eidx  = (const int*)d_in[5];
  const int*   batch = (const int*)d_in[6];

  const int nN = in_sizes[0] / D;   // 100000
  const int nE = in_sizes[5] / 2;   // 1600000
  const int* src = eidx;            // edge_index[0]
  const int* dst = eidx + nE;       // edge_index[1]

  // workspace layout (floats): buf1 | buf2 | dinv | sums(16*64) | cnts(16)
  float* buf1 = (float*)d_ws;
  float* buf2 = buf1 + (size_t)nN * D;
  float* dinv = buf2 + (size_t)nN * D;
  float* sums = dinv + nN;
  float* cnts = sums + 16 * D;

  const int T = 256;
  auto cdiv = [](long long a, long long b) { return (int)((a + b - 1) / b); };
  const int totND = nN * D;                 // 6.4M
  const int totED = nE * D;                 // 102.4M

  // degrees with self-loops -> dinv = deg^-1/2
  k_fill<<<cdiv(nN, T), T, 0, stream>>>(dinv, 1.0f, nN);
  k_deg_accum<<<cdiv(nE, T), T, 0, stream>>>(dst, dinv, nE);
  k_rsqrt_inplace<<<cdiv(nN, T), T, 0, stream>>>(dinv, nN);

  const int gemmBlocks = cdiv(nN, 32);      // 3125; nN divisible by 32

  // ---- layer 1 ----
  k_gemm_wmma<<<gemmBlocks, T, 0, stream>>>(x, W1, buf1, nN);
  k_selfloop<<<cdiv(totND, T), T, 0, stream>>>(buf1, dinv, buf2, totND);
  k_edge_scatter<<<cdiv(totED, T), T, 0, stream>>>(buf1, dinv, src, dst, buf2, nE);
  k_bias_act<<<cdiv(totND, T), T, 0, stream>>>(buf2, b1, totND, 1);

  // ---- layer 2 ----
  k_gemm_wmma<<<gemmBlocks, T, 0, stream>>>(buf2, W2, buf1, nN);
  k_selfloop<<<cdiv(totND, T), T, 0, stream>>>(buf1, dinv, buf2, totND);
  k_edge_scatter<<<cdiv(totED, T), T, 0, stream>>>(buf1, dinv, src, dst, buf2, nE);
  k_bias_act<<<cdiv(totND, T), T, 0, stream>>>(buf2, b2, totND, 0);

  // ---- global mean pool ----
  k_fill<<<cdiv(16 * D + 16, T), T, 0, stream>>>(sums, 0.0f, 16 * D + 16); // sums+cnts contiguous
  k_pool_accum<<<cdiv(totND, T), T, 0, stream>>>(buf2, batch, sums, cnts, totND);
  k_pool_div<<<cdiv(16 * D, T), T, 0, stream>>>(sums, cnts, (float*)d_out, 16 * D);
}